// PropagLayer_21912923144682
// MI455X (gfx1250) — compile-verified
//
#include <hip/hip_runtime.h>

typedef __attribute__((ext_vector_type(4))) unsigned int tdm_v4u;
typedef __attribute__((ext_vector_type(8))) int          tdm_v8i;
typedef __attribute__((ext_vector_type(4))) int          tdm_v4i;

#define PROPAG_K 64
#define LOG2E_F   1.4426950408889634f
#define INV2PI_F  0.15915494309189535f

__global__ __launch_bounds__(256) void propag_kernel(
    const float* __restrict__ params,   // [K,3] rows: {Gamma, omega, gamma}
    const float* __restrict__ tgrid,    // [T]
    float* __restrict__ out,            // [T]
    int T)
{
    __shared__ float sRaw[PROPAG_K * 3];   // raw parameter block (TDM target)
    __shared__ float sGam[PROPAG_K];       // Gamma_k
    __shared__ float sNgl[PROPAG_K];       // -gamma_k * log2(e)
    __shared__ float sWrv[PROPAG_K];       // omega_k / (2*pi)

    const int tid = threadIdx.x;

#if __has_builtin(__builtin_amdgcn_tensor_load_to_lds)
    // --- Stage the 768B parameter block into LDS with the Tensor Data Mover.
    // Wave-uniform guard: only wave 0 issues the DMA (TDM ignores EXEC,
    // TENSORcnt is per-wave). D# per CDNA5 ISA ch.8: 1D tile of 192 f32.
    if (tid < 32) {
        const unsigned           ldsAddr = (unsigned)(size_t)(&sRaw[0]); // low 32b of flat = LDS offset
        const unsigned long long ga      = (unsigned long long)(size_t)params;
        tdm_v4u g0;
        g0.x = 1u;                                        // count=1, is_restore=0, gather off
        g0.y = ldsAddr;                                   // lds_addr
        g0.z = (unsigned)(ga & 0xFFFFFFFFull);            // global_addr[31:0]
        g0.w = (unsigned)((ga >> 32) & 0x01FFFFFFull)     // global_addr[56:32]
             | (2u << 30);                                // type = 2 ("image")
        tdm_v8i g1;
        g1[0] = (int)(2u << 16);                          // data_size = 2 -> 4 bytes
        g1[1] = (int)((unsigned)(PROPAG_K * 3) << 16);    // tensor_dim0 = 192 (bits 79:48, lo half)
        g1[2] = (int)(1u << 16);                          // tensor_dim0 hi = 0; tensor_dim1 = 1
        g1[3] = (int)((unsigned)(PROPAG_K * 3) << 16);    // tensor_dim1 hi = 0; tile_dim0 = 192
        g1[4] = 1;                                        // tile_dim1 = 1, tile_dim2 = 0
        g1[5] = PROPAG_K * 3;                             // tensor_dim0_stride = 192
        g1[6] = 0;                                        // stride hi / tensor_dim1_stride lo
        g1[7] = 0;
        tdm_v4i gz4 = {0, 0, 0, 0};                       // groups 2/3 unused (<=2D tensor)
        tdm_v8i gz8 = {0, 0, 0, 0, 0, 0, 0, 0};          // 6-arg (clang-23) extra operand
        __builtin_amdgcn_tensor_load_to_lds(g0, g1, gz4, gz4, gz8, 0);
        __builtin_amdgcn_s_wait_tensorcnt(0);             // s_wait_tensorcnt 0
    }
#else
    for (int i = tid; i < PROPAG_K * 3; i += (int)blockDim.x) sRaw[i] = params[i];
#endif
    __syncthreads();

    // Derive per-oscillator constants so the hot loop needs exactly one
    // v_exp_f32 and one v_cos_f32 per (t,k) pair (args pre-scaled).
    if (tid < PROPAG_K) {
        const float G = sRaw[3 * tid + 0];
        const float w = sRaw[3 * tid + 1];
        const float g = sRaw[3 * tid + 2];
        sGam[tid] = G;
        sNgl[tid] = -g * LOG2E_F;   // e^{-g t} = 2^{(-g*log2 e) t}
        sWrv[tid] =  w * INV2PI_F;  // v_cos_f32 input is in revolutions
    }
    __syncthreads();

    // 4 time points per lane: 128-bit global load/store, 4 independent
    // accumulators to keep the transcendental pipe saturated.
    const size_t base = ((size_t)blockIdx.x * blockDim.x + (size_t)tid) * 4;
    if (base + 4 <= (size_t)T) {
        const float4 t4 = *reinterpret_cast<const float4*>(tgrid + base);
        float a0 = 0.f, a1 = 0.f, a2 = 0.f, a3 = 0.f;
        #pragma unroll 8
        for (int k = 0; k < PROPAG_K; ++k) {
            const float G   = sGam[k];
            const float ngl = sNgl[k];
            const float wrv = sWrv[k];
            const float e0 = __builtin_amdgcn_exp2f(ngl * t4.x);
            const float e1 = __builtin_amdgcn_exp2f(ngl * t4.y);
            const float e2 = __builtin_amdgcn_exp2f(ngl * t4.z);
            const float e3 = __builtin_amdgcn_exp2f(ngl * t4.w);
            const float c0 = __builtin_amdgcn_cosf(wrv * t4.x);
            const float c1 = __builtin_amdgcn_cosf(wrv * t4.y);
            const float c2 = __builtin_amdgcn_cosf(wrv * t4.z);
            const float c3 = __builtin_amdgcn_cosf(wrv * t4.w);
            a0 = __builtin_fmaf(G * e0, c0, a0);
            a1 = __builtin_fmaf(G * e1, c1, a1);
            a2 = __builtin_fmaf(G * e2, c2, a2);
            a3 = __builtin_fmaf(G * e3, c3, a3);
        }
        float4 r; r.x = a0; r.y = a1; r.z = a2; r.w = a3;
        *reinterpret_cast<float4*>(out + base) = r;
    } else {
        // Scalar tail (T not divisible by 1024) — never taken for T = 2^21.
        for (size_t i = base; i < (size_t)T; ++i) {
            const float t = tgrid[i];
            float a = 0.f;
            for (int k = 0; k < PROPAG_K; ++k) {
                a = __builtin_fmaf(sGam[k] * __builtin_amdgcn_exp2f(sNgl[k] * t),
                                   __builtin_amdgcn_cosf(sWrv[k] * t), a);
            }
            out[i] = a;
        }
    }
}

extern "C" void kernel_launch(void* const* d_in, const int* in_sizes, int n_in,
                              void* d_out, int out_size, void* d_ws, size_t ws_size,
                              hipStream_t stream)
{
    const float* params = (const float*)d_in[0];  // [64,3] float32
    const float* tgrid  = (const float*)d_in[1];  // [T] float32
    float*       out    = (float*)d_out;          // [T] float32
    const int T = in_sizes[1];
    const int perBlock = 256 * 4;
    const int blocks = (T + perBlock - 1) / perBlock;
    propag_kernel<<<blocks, 256, 0, stream>>>(params, tgrid, out, T);
}